// OneHotMLP_73813307949089
// MI455X (gfx1250) — compile-verified
//
#include <hip/hip_runtime.h>
#include <hip/hip_bf16.h>

// Sizes (compile-time constants from the reference)
#define S_DIM 4096
#define P_DIM 256
#define V_DIM 33
#define D_DIM 1024
#define CVT   (P_DIM * V_DIM)   // 8448 = one-hot flattened width

typedef __attribute__((ext_vector_type(2))) float    v2f;
typedef __attribute__((ext_vector_type(8))) float    v8f;
typedef __attribute__((ext_vector_type(4))) unsigned v4u;
typedef __attribute__((ext_vector_type(4))) int      v4i;
typedef __attribute__((ext_vector_type(8))) int      v8i;

// ---------------------------------------------------------------------------
// TDM: async-load a 2D tile of 128 rows x 16 f32 (row stride 1024 f32) from
// global memory into LDS, with 1-DWORD padding after every 16 DWORDs so each
// LDS row is 17 floats (bank-conflict-free WMMA fragment reads).
// Descriptor per CDNA5 ISA 8.3/8.4; groups 2/3 zero (2D tensor).
// This toolchain exposes the 6-arg builtin form:
//   (uint32x4 g0, int32x8 g1, int32x4 g2, int32x4 g3, int32x8 pad, i32 cpol)
// ---------------------------------------------------------------------------
__device__ __forceinline__ void tdm_load_tile_128x16(const float* gsrc,
                                                     unsigned lds_byte_off) {
  const unsigned long long ga = (unsigned long long)gsrc;
  v4u g0;
  g0.x = 1u;                                   // count=1, user descriptor
  g0.y = lds_byte_off;                         // lds_addr (bytes)
  g0.z = (unsigned)ga;                         // global_addr[31:0]
  g0.w = (unsigned)((ga >> 32) & 0x1FFFFFFull) // global_addr[56:32]
         | (2u << 30);                         // type=2 ("image")
  v8i g1;
  g1[0] = (int)((2u << 16)      // data_size = 4 bytes
              | (1u << 20)      // pad_enable
              | (3u << 22)      // pad_interval: every 16 DWORDs
              | (0u << 25));    // pad_amount: 1 DWORD
  g1[1] = (int)(1024u << 16);   // tensor_dim0[15:0] = 1024 (bits 63:48)
  g1[2] = (int)(128u << 16);    // tensor_dim0 hi = 0; tensor_dim1 = 128
  g1[3] = (int)(16u << 16);     // tensor_dim1 hi = 0; tile_dim0 = 16
  g1[4] = 128;                  // tile_dim1 = 128; tile_dim2 = 0 (2D)
  g1[5] = 1024;                 // tensor_dim0_stride[31:0] = 1024 elements
  g1[6] = 0;                    // stride0 hi, dim1_stride lo
  g1[7] = 0;                    // dim1_stride hi
  const v4i z4 = {};
  const v8i z8 = {};
  __builtin_amdgcn_tensor_load_to_lds(g0, g1, z4, z4, z8, 0);
}

// ---------------------------------------------------------------------------
// Kernel 1: transpose W1 [D, P*V] -> W1T [P*V, D] so layer-1 gathers read
// contiguous 4KB rows (coalesced) instead of stride-8448 columns.
// ---------------------------------------------------------------------------
__global__ __launch_bounds__(256) void w1_transpose_kernel(
    const float* __restrict__ W1, float* __restrict__ W1T) {
  __shared__ float tile[32][33];
  const int c = blockIdx.x * 32 + threadIdx.x;   // column in W1 (0..8447)
  const int d = blockIdx.y * 32 + threadIdx.y;   // row in W1    (0..1023)
#pragma unroll
  for (int j = 0; j < 32; j += 8)
    tile[threadIdx.y + j][threadIdx.x] = W1[(size_t)(d + j) * CVT + c];
  __syncthreads();
  const int od = blockIdx.y * 32 + threadIdx.x;  // col in W1T (d)
  const int oc = blockIdx.x * 32 + threadIdx.y;  // row in W1T (c)
#pragma unroll
  for (int j = 0; j < 32; j += 8)
    W1T[(size_t)(oc + j) * D_DIM + od] = tile[threadIdx.x][threadIdx.y + j];
}

// ---------------------------------------------------------------------------
// Kernel 2: layer 1 as embedding-bag gather-sum (one-hot GEMM collapsed).
// h1[s,d] = relu(b1[d] + sum_p W1T[p*33 + x[s,p]][d]); W1T rows live in L2.
// ---------------------------------------------------------------------------
__global__ __launch_bounds__(256) void layer1_gather_kernel(
    const int* __restrict__ x, const float* __restrict__ W1T,
    const float* __restrict__ b1, float* __restrict__ out) {
  __shared__ int idx[P_DIM];
  const int s = blockIdx.x;
  const int t = threadIdx.x;
  idx[t] = x[(size_t)s * P_DIM + t];
  __syncthreads();
  const int d0 = t * 4;
  float4 acc = *(const float4*)(b1 + d0);
#pragma unroll 4
  for (int p = 0; p < P_DIM; ++p) {
    const int c = p * V_DIM + idx[p];
    if (p + 8 < P_DIM) {
      const int cf = (p + 8) * V_DIM + idx[p + 8];
      __builtin_prefetch(W1T + (size_t)cf * D_DIM + d0, 0, 0);  // global_prefetch_b8
    }
    const float4 w = *(const float4*)(W1T + (size_t)c * D_DIM + d0);
    acc.x += w.x; acc.y += w.y; acc.z += w.z; acc.w += w.w;
  }
  acc.x = fmaxf(acc.x, 0.f);
  acc.y = fmaxf(acc.y, 0.f);
  acc.z = fmaxf(acc.z, 0.f);
  acc.w = fmaxf(acc.w, 0.f);
  *(float4*)(out + (size_t)s * D_DIM + d0) = acc;
}

// ---------------------------------------------------------------------------
// Kernel 3: f32 WMMA GEMM, TDM double-buffered tiles, fused bias (+ReLU).
// out[m,n] = act( sum_k A[m,k] * W[n,k] + bias[n] )     (torch W convention)
//
// Workgroup = 256 threads (8 waves) -> 128(M) x 128(N) tile.
// Wave (wm = wave&3 -> 32 M-rows, wn = wave>>2 -> 64 N-cols): 2x4 grid of
// 16x16 accumulators (8 x v8f = 64 VGPRs).
//
// Both A and W tiles are loaded UNtransposed by TDM as [row][k] with padded
// 17-float rows; fragment reads (ISA 7.12.2, 32-bit, wave32):
//   A lane L, vgpr j : As[m][2*(L>>4)+j]      (m = row)
//   B lane L, vgpr j : Ws[n][2*(L>>4)+j]      (B[k][n] == W[n][k])
//   C/D lane L, vgpr v: row = v + 8*(L>>4), col = L&15
// Column accesses stride 17 -> conflict-free across the 16-lane groups.
// ---------------------------------------------------------------------------
#define BM 128
#define BN 128
#define TK 16
#define LROW 17   // padded LDS row length in floats

__global__ __launch_bounds__(256) void gemm_bias_act_kernel(
    const float* __restrict__ A,    // [S_DIM, D_DIM], already ReLU'd
    const float* __restrict__ W,    // [D_DIM, D_DIM] row-major [out,in]
    const float* __restrict__ bias, // [D_DIM]
    float* __restrict__ out,        // [S_DIM, D_DIM]
    int relu_out) {
  __shared__ float As[2][BM * LROW];
  __shared__ float Ws[2][BN * LROW];

  const int tid  = threadIdx.x;
  const int wave = tid >> 5;
  const int lane = tid & 31;
  const int lo16 = lane & 15;
  const int hi   = lane >> 4;
  const int wm   = wave & 3;   // 32 M-rows per wave
  const int wn   = wave >> 2;  // 64 N-cols per wave
  const int m0   = blockIdx.y * BM;
  const int n0   = blockIdx.x * BN;

  v8f acc[2][4] = {};

  // Scalar (SALU) leader test so non-leader waves *branch around* the TDM
  // issue (tensor ops ignore EXEC, a masked path would duplicate the DMA).
  const bool leader = (__builtin_amdgcn_readfirstlane((int)(tid >> 5)) == 0);

  const float* Abase = A + (size_t)m0 * D_DIM;
  const float* Wbase = W + (size_t)n0 * D_DIM;

  if (leader) {
    tdm_load_tile_128x16(Abase, (unsigned)(size_t)&As[0][0]);
    tdm_load_tile_128x16(Wbase, (unsigned)(size_t)&Ws[0][0]);
    __builtin_amdgcn_s_wait_tensorcnt(0);
  }
  __syncthreads();

  for (int kt = 0; kt < D_DIM; kt += TK) {
    const int cur = (kt >> 4) & 1;
    // Prefetch next K-tile into the other buffer (overlaps with compute).
    if (leader && (kt + TK < D_DIM)) {
      tdm_load_tile_128x16(Abase + kt + TK, (unsigned)(size_t)&As[cur ^ 1][0]);
      tdm_load_tile_128x16(Wbase + kt + TK, (unsigned)(size_t)&Ws[cur ^ 1][0]);
    }
    const float* as = &As[cur][0];
    const float* ws = &Ws[cur][0];

#pragma unroll
    for (int kk = 0; kk < TK; kk += 4) {
      v2f a[2], b[4];
#pragma unroll
      for (int mm = 0; mm < 2; ++mm) {
        const int r = wm * 32 + mm * 16 + lo16;
        a[mm].x = as[r * LROW + kk + 2 * hi + 0];
        a[mm].y = as[r * LROW + kk + 2 * hi + 1];
      }
#pragma unroll
      for (int nb = 0; nb < 4; ++nb) {
        const int r = wn * 64 + nb * 16 + lo16;
        b[nb].x = ws[r * LROW + kk + 2 * hi + 0];
        b[nb].y = ws[r * LROW + kk + 2 * hi + 1];
      }
#pragma unroll
      for (int mm = 0; mm < 2; ++mm)
#pragma unroll
        for (int nb = 0; nb < 4; ++nb)
          acc[mm][nb] = __builtin_amdgcn_wmma_f32_16x16x4_f32(
              /*neg_a=*/false, a[mm], /*neg_b=*/false, b[nb],
              /*c_mod=*/(short)0, acc[mm][nb],
              /*reuse_a=*/false, /*reuse_b=*/false);
    }

    if (leader) __builtin_amdgcn_s_wait_tensorcnt(0);
    __syncthreads();
  }

  // Epilogue: bias + optional ReLU, per C/D VGPR layout.
#pragma unroll
  for (int nb = 0; nb < 4; ++nb) {
    const int n  = n0 + wn * 64 + nb * 16 + lo16;
    const float bv = bias[n];
#pragma unroll
    for (int mm = 0; mm < 2; ++mm) {
#pragma unroll
      for (int v = 0; v < 8; ++v) {
        const int m = m0 + wm * 32 + mm * 16 + v + 8 * hi;
        float val = acc[mm][nb][v] + bv;
        if (relu_out) val = fmaxf(val, 0.f);
        out[(size_t)m * D_DIM + n] = val;
      }
    }
  }
}

// ---------------------------------------------------------------------------
// Kernel 4: output layer (O=1): y[s] = H[s,:] . Wo + bo
// ---------------------------------------------------------------------------
__global__ __launch_bounds__(256) void out_layer_kernel(
    const float* __restrict__ H, const float* __restrict__ Wo,
    const float* __restrict__ bo, float* __restrict__ y) {
  __shared__ float red[256];
  const int s = blockIdx.x;
  const int t = threadIdx.x;
  const float4 h = *(const float4*)(H + (size_t)s * D_DIM + t * 4);
  const float4 w = *(const float4*)(Wo + t * 4);
  red[t] = h.x * w.x + h.y * w.y + h.z * w.z + h.w * w.w;
  __syncthreads();
#pragma unroll
  for (int off = 128; off > 0; off >>= 1) {
    if (t < off) red[t] += red[t + off];
    __syncthreads();
  }
  if (t == 0) y[s] = red[0] + bo[0];
}

// ---------------------------------------------------------------------------
extern "C" void kernel_launch(void* const* d_in, const int* in_sizes, int n_in,
                              void* d_out, int out_size, void* d_ws, size_t ws_size,
                              hipStream_t stream) {
  const int*   x  = (const int*)d_in[0];
  const float* W1 = (const float*)d_in[1];
  const float* b1 = (const float*)d_in[2];
  const float* W2 = (const float*)d_in[3];
  const float* b2 = (const float*)d_in[4];
  const float* W3 = (const float*)d_in[5];
  const float* b3 = (const float*)d_in[6];
  const float* Wo = (const float*)d_in[7];
  const float* bo = (const float*)d_in[8];
  float* y = (float*)d_out;

  float* ws  = (float*)d_ws;
  float* W1T = ws;                                    // 8448 * 1024 floats
  float* hA  = W1T + (size_t)CVT * D_DIM;             // 4096 * 1024 floats
  float* hB  = hA + (size_t)S_DIM * D_DIM;            // 4096 * 1024 floats

  // 1) Transpose W1 for coalesced gathers.
  w1_transpose_kernel<<<dim3(CVT / 32, D_DIM / 32), dim3(32, 8), 0, stream>>>(W1, W1T);

  // 2) Layer 1: embedding-bag gather; fused bias + ReLU.
  layer1_gather_kernel<<<S_DIM, 256, 0, stream>>>(x, W1T, b1, hA);

  // 3) Layers 2 & 3: f32-WMMA GEMMs with TDM-fed LDS tiles; bias + ReLU.
  dim3 ggrid(D_DIM / BN, S_DIM / BM);  // (8, 32)
  gemm_bias_act_kernel<<<ggrid, 256, 0, stream>>>(hA, W2, b2, hB, 1);
  gemm_bias_act_kernel<<<ggrid, 256, 0, stream>>>(hB, W3, b3, hA, 1);

  // 4) Output layer (no ReLU on y).
  out_layer_kernel<<<S_DIM, 256, 0, stream>>>(hA, Wo, bo, y);
}